// NodesAggregation_46325517254752
// MI455X (gfx1250) — compile-verified
//
#include <hip/hip_runtime.h>

typedef __attribute__((ext_vector_type(2))) float v2f;
typedef __attribute__((ext_vector_type(8))) float v8f;

namespace {
constexpr int L = 5, P = 5, T = 15, C = 8;
constexpr int N = 1024, E = 8192;
constexpr int PC = P * C;                         // 40
constexpr int FROW = L * PC;                      // 200 floats per (l, node)
constexpr int BES = P * L * L;                    // 125 floats per (l1,l2,e)
constexpr size_t OUT_HALF = (size_t)L * N * T * L * PC;  // 15,360,000

union V8 { v8f v; float f[8]; };

// D(16x16) = sum_{kk<ksteps} A(16x4) @ B(4x16), A/B row-major in LDS.
// f32 WMMA lane layout (ISA 7.12.2): lanes 0-15 carry M/N, half-wave selects K pair.
__device__ inline v8f wmma_tile(const float* A, int lda,
                                const float* B, int ldb,
                                int ksteps, int lane) {
  const int h  = lane >> 4;   // 0: lanes 0-15, 1: lanes 16-31
  const int lm = lane & 15;
  v8f acc = {};
  for (int kk = 0; kk < ksteps; ++kk) {
    const int kb = kk * 4 + h * 2;
    v2f a, b;
    a.x = A[lm * lda + kb + 0];
    a.y = A[lm * lda + kb + 1];
    b.x = B[(kb + 0) * ldb + lm];
    b.y = B[(kb + 1) * ldb + lm];
    acc = __builtin_amdgcn_wmma_f32_16x16x4_f32(false, a, false, b,
                                                (short)0, acc, false, false);
  }
  return acc;
}

// Async copy of one bessel (l1,l2,e) block (125 f32, contiguous) into LDS via the
// async engine (GLOBAL_LOAD_ASYNC_TO_LDS_B32, tracked by ASYNCcnt).
__device__ inline void issue_bessel_async(const float* __restrict__ bes_r,
                                          const float* __restrict__ bes_i,
                                          int e, int idx,
                                          float* dstR, float* dstI, int lane) {
  const size_t off = ((size_t)idx * E + e) * BES;
  for (int i = lane; i < BES; i += 32) {
    unsigned lr = (unsigned)(size_t)(dstR + i);          // low 32 bits = LDS offset
    unsigned li = (unsigned)(size_t)(dstI + i);
    unsigned long long gr = (unsigned long long)(size_t)(bes_r + off + i);
    unsigned long long gi = (unsigned long long)(size_t)(bes_i + off + i);
    asm volatile("global_load_async_to_lds_b32 %0, %1, off"
                 :: "v"(lr), "v"(gr) : "memory");
    asm volatile("global_load_async_to_lds_b32 %0, %1, off"
                 :: "v"(li), "v"(gi) : "memory");
  }
}

__device__ inline void wait_async0() {
  asm volatile("s_wait_asynccnt 0x0" ::: "memory");
}
} // namespace

// out[l2,n,t,a,p,c] = w[t] * mask(a<=l2) * feat[l2,n,a,p,c]   (both halves)
__global__ void __launch_bounds__(256)
init_out_kernel(const float* __restrict__ feat_r,
                const float* __restrict__ feat_i,
                const float* __restrict__ w,
                float* __restrict__ out) {
  size_t idx = (size_t)blockIdx.x * blockDim.x + threadIdx.x;
  if (idx >= OUT_HALF) return;
  int pc = (int)(idx % PC); size_t r = idx / PC;
  int a  = (int)(r % L);  r /= L;
  int t  = (int)(r % T);  r /= T;
  int n  = (int)(r % N);  int l2 = (int)(r / N);
  float wv = (a <= l2) ? w[t] : 0.f;
  size_t fidx = (((size_t)l2 * N + n) * L + a) * PC + pc;
  out[idx]            = wv * feat_r[fidx];
  out[OUT_HALF + idx] = wv * feat_i[fidx];
}

// One wave (32 lanes) per edge; all small matmuls via V_WMMA_F32_16X16X4_F32,
// bessel stream double-buffered in LDS via async global->LDS loads.
__global__ void __launch_bounds__(32)
edge_kernel(const float* __restrict__ feat_r, const float* __restrict__ feat_i,
            const float* __restrict__ fsm_pr, const float* __restrict__ fsm_pi,
            const float* __restrict__ fsm_nr, const float* __restrict__ fsm_ni,
            const float* __restrict__ bes_r,  const float* __restrict__ bes_i,
            const float* __restrict__ ssm_pr, const float* __restrict__ ssm_pi,
            const float* __restrict__ ssm_nr, const float* __restrict__ ssm_ni,
            const int* __restrict__ src_idx, const int* __restrict__ dst_idx,
            const int* __restrict__ et_idx,  float* __restrict__ out) {
  __shared__ float xcat[12 * 48];   // K=12 (rr|ri|pad) x N=48 (40+pad)
  __shared__ float acR[16 * 12];    // [pr+s*nr | s*ni-pi]
  __shared__ float acI[16 * 12];    // [pi+s*ni | pr-s*nr]
  __shared__ float rotr[L * PC];
  __shared__ float roti[L * PC];
  __shared__ float frtr[L * FROW];
  __shared__ float frti[L * FROW];
  __shared__ float bA [16 * 32];    // block-diag [br|bi] for up to 3 p-batches
  __shared__ float bBr[32 * 16];    // stacked [tr;ti]
  __shared__ float bBi[32 * 16];    // stacked [ti;-tr]
  __shared__ float stgR[2][128];    // async-staged bessel_real block (ping-pong)
  __shared__ float stgI[2][128];    // async-staged bessel_imag block

  const int e    = blockIdx.x;
  const int lane = threadIdx.x;
  const int h    = lane >> 4;
  const int lm   = lane & 15;

  const int src = src_idx[e];
  const int dst = dst_idx[e];
  const int et  = et_idx[e];

  // Kick off the async stream for bessel block (l1=0,l2=0) immediately.
  issue_bessel_async(bes_r, bes_i, e, 0, stgR[0], stgI[0], lane);

  for (int i = lane; i < L * FROW; i += 32) { frtr[i] = 0.f; frti[i] = 0.f; }
  __syncthreads();

  for (int l1 = 0; l1 < L; ++l1) {
    const float s1 = (l1 & 1) ? -1.f : 1.f;

    if (l1 + 1 < L && lane < 13) {  // stream-ahead next fsm matrices (gfx1250 prefetch)
      const size_t nmb = ((size_t)(l1 + 1) * E + e) * (L * L);
      __builtin_prefetch(fsm_pr + nmb + lane * 2, 0, 0);
      __builtin_prefetch(fsm_pi + nmb + lane * 2, 0, 0);
      __builtin_prefetch(fsm_nr + nmb + lane * 2, 0, 0);
      __builtin_prefetch(fsm_ni + nmb + lane * 2, 0, 0);
    }

    // ---------- build Xcat: rows 0-4 = rr (masked l'<=l1), rows 5-9 = ri ----------
    for (int i = lane; i < 12 * 48; i += 32) xcat[i] = 0.f;
    for (int i = lane; i < 16 * 12; i += 32) { acR[i] = 0.f; acI[i] = 0.f; }
    __syncthreads();
    {
      const float* fr = feat_r + ((size_t)l1 * N + src) * FROW;
      const float* fi = feat_i + ((size_t)l1 * N + src) * FROW;
      for (int i = lane; i < 2 * L * PC; i += 32) {      // 400 entries
        int row = i / PC, col = i % PC;
        int lp = row % L;
        float v = 0.f;
        if (lp <= l1) v = (row < L ? fr : fi)[lp * PC + col];
        xcat[row * 48 + col] = v;
      }
      const size_t mb = ((size_t)l1 * E + e) * (L * L);
      for (int i = lane; i < L * L; i += 32) {
        int a = i / L, b = i % L;
        float pr = fsm_pr[mb + i], pi = fsm_pi[mb + i];
        float nr = fsm_nr[mb + i], ni = fsm_ni[mb + i];
        acR[a * 12 + b]     = pr + s1 * nr;
        acR[a * 12 + 5 + b] = s1 * ni - pi;
        acI[a * 12 + b]     = pi + s1 * ni;
        acI[a * 12 + 5 + b] = pr - s1 * nr;
      }
    }
    __syncthreads();

    // ---------- rot(5x40) = Acat(5x10) @ Xcat(10x40), 3 N-tiles, K-steps=3 ----------
    for (int nt = 0; nt < 3; ++nt) {
      V8 dr, di;
      dr.v = wmma_tile(acR, 12, xcat + nt * 16, 48, 3, lane);
      di.v = wmma_tile(acI, 12, xcat + nt * 16, 48, 3, lane);
      const int col = nt * 16 + lm;
      if (h == 0 && col < PC) {
#pragma unroll
        for (int r = 0; r < L; ++r) {
          rotr[r * PC + col] = (r <= l1) ? dr.f[r] : 0.f;
          roti[r * PC + col] = (r <= l1) ? di.f[r] : 0.f;
        }
      }
    }
    __syncthreads();

    // ---------- bessel contraction, block-diag 3 (then 2) p-batches per tile ----------
    for (int l2 = 0; l2 < L; ++l2) {
      const int idx = l1 * L + l2;
      const int cur = idx & 1;

      wait_async0();        // staged (l1,l2) block resident in LDS
      __syncthreads();
      if (idx + 1 < L * L)  // overlap next block's DMA with this block's WMMAs
        issue_bessel_async(bes_r, bes_i, e, idx + 1,
                           stgR[cur ^ 1], stgI[cur ^ 1], lane);

      const float* br = stgR[cur];
      const float* bi = stgI[cur];

      for (int g = 0; g < 2; ++g) {
        const int pcnt = g ? 2 : 3;     // p in {0,1,2} then {3,4}
        const int ks   = g ? 5 : 8;     // ceil(pcnt*10 / 4)
        for (int i = lane; i < 16 * 32; i += 32) bA[i] = 0.f;
        for (int i = lane; i < 32 * 16; i += 32) { bBr[i] = 0.f; bBi[i] = 0.f; }
        __syncthreads();
        for (int i = lane; i < pcnt * 25; i += 32) {
          int j = i / 25, ab = i % 25;
          int a = ab / 5, b = ab % 5;
          bA[(j * 5 + a) * 32 + j * 10 + b]     = br[(g * 3 + j) * 25 + ab];
          bA[(j * 5 + a) * 32 + j * 10 + 5 + b] = bi[(g * 3 + j) * 25 + ab];
        }
        for (int i = lane; i < pcnt * 10 * C; i += 32) {
          int j = i / (10 * C), rem = i % (10 * C);
          int rb = rem / C, c = rem % C;
          int p = g * 3 + j;
          int b = (rb >= 5) ? rb - 5 : rb;
          float trv = rotr[b * PC + p * C + c];
          float tiv = roti[b * PC + p * C + c];
          bBr[(j * 10 + rb) * 16 + c] = (rb >= 5) ? tiv : trv;   // [tr; ti]
          bBi[(j * 10 + rb) * 16 + c] = (rb >= 5) ? -trv : tiv;  // [ti;-tr]
        }
        __syncthreads();
        V8 ddr, ddi;
        ddr.v = wmma_tile(bA, 32, bBr, 16, ks, lane);   // dd_r = br@tr + bi@ti
        ddi.v = wmma_tile(bA, 32, bBi, 16, ks, lane);   // dd_i = br@ti - bi@tr
        const int c = lm;
        if (c < C) {
#pragma unroll
          for (int r = 0; r < 8; ++r) {
            int row = r + 8 * h;
            if (row < pcnt * 5) {
              int j = row / 5, a = row % 5;
              int p = g * 3 + j;
              frtr[l2 * FROW + a * PC + p * C + c] += ddr.f[r];
              frti[l2 * FROW + a * PC + p * C + c] += ddi.f[r];
            }
          }
        }
        __syncthreads();
      }
    }
  } // l1

  // ---------- ssm rotation + segment-sum via global f32 atomics ----------
  for (int l2 = 0; l2 < L; ++l2) {
    const float s2 = (l2 & 1) ? -1.f : 1.f;
    for (int i = lane; i < 12 * 48; i += 32) xcat[i] = 0.f;
    for (int i = lane; i < 16 * 12; i += 32) { acR[i] = 0.f; acI[i] = 0.f; }
    __syncthreads();
    for (int i = lane; i < 2 * L * PC; i += 32) {
      int row = i / PC, col = i % PC;
      int lp = row % L;
      float v = 0.f;
      if (lp <= l2) v = (row < L ? frtr : frti)[l2 * FROW + lp * PC + col];
      xcat[row * 48 + col] = v;
    }
    const size_t mb = ((size_t)l2 * E + e) * (L * L);
    for (int i = lane; i < L * L; i += 32) {
      int a = i / L, b = i % L;
      float pr = ssm_pr[mb + i], pi = ssm_pi[mb + i];
      float nr = ssm_nr[mb + i], ni = ssm_ni[mb + i];
      acR[a * 12 + b]     = pr + s2 * nr;
      acR[a * 12 + 5 + b] = s2 * ni - pi;
      acI[a * 12 + b]     = pi + s2 * ni;
      acI[a * 12 + 5 + b] = pr - s2 * nr;
    }
    __syncthreads();
    float* outR = out + (((size_t)l2 * N + dst) * T + et) * (L * PC);
    float* outI = outR + OUT_HALF;
    for (int nt = 0; nt < 3; ++nt) {
      V8 dr, di;
      dr.v = wmma_tile(acR, 12, xcat + nt * 16, 48, 3, lane);
      di.v = wmma_tile(acI, 12, xcat + nt * 16, 48, 3, lane);
      const int col = nt * 16 + lm;
      if (h == 0 && col < PC) {
#pragma unroll
        for (int r = 0; r < L; ++r) {
          if (r <= l2) {
            atomicAdd(&outR[r * PC + col], dr.f[r]);
            atomicAdd(&outI[r * PC + col], di.f[r]);
          }
        }
      }
    }
    __syncthreads();
  }
}

extern "C" void kernel_launch(void* const* d_in, const int* in_sizes, int n_in,
                              void* d_out, int out_size, void* d_ws, size_t ws_size,
                              hipStream_t stream) {
  (void)in_sizes; (void)n_in; (void)out_size; (void)d_ws; (void)ws_size;
  const float* feat_r = (const float*)d_in[0];
  const float* feat_i = (const float*)d_in[1];
  const float* fsm_pr = (const float*)d_in[2];
  const float* fsm_pi = (const float*)d_in[3];
  const float* fsm_nr = (const float*)d_in[4];
  const float* fsm_ni = (const float*)d_in[5];
  const float* bes_r  = (const float*)d_in[6];
  const float* bes_i  = (const float*)d_in[7];
  const float* ssm_pr = (const float*)d_in[8];
  const float* ssm_pi = (const float*)d_in[9];
  const float* ssm_nr = (const float*)d_in[10];
  const float* ssm_ni = (const float*)d_in[11];
  const float* w      = (const float*)d_in[12];
  const int*   src    = (const int*)d_in[13];
  const int*   dst    = (const int*)d_in[14];
  const int*   et     = (const int*)d_in[15];
  float* out = (float*)d_out;

  const int init_blocks = (int)((OUT_HALF + 255) / 256);
  init_out_kernel<<<init_blocks, 256, 0, stream>>>(feat_r, feat_i, w, out);
  edge_kernel<<<E, 32, 0, stream>>>(feat_r, feat_i,
                                    fsm_pr, fsm_pi, fsm_nr, fsm_ni,
                                    bes_r, bes_i,
                                    ssm_pr, ssm_pi, ssm_nr, ssm_ni,
                                    src, dst, et, out);
}